// HyperGAT_64570538328499
// MI455X (gfx1250) — compile-verified
//
#include <hip/hip_runtime.h>
#include <math.h>

typedef __attribute__((ext_vector_type(2))) float v2f;
typedef __attribute__((ext_vector_type(8))) float v8f;

#define NEG_SLOPE 0.2f

// ---------------------------------------------------------------------------
// WMMA f32 GEMM: C[M,N] = A[M,K] * B[K,N], row-major, M%16==N%16==K%4==0.
// One wave per 16x16 output tile; V_WMMA_F32_16X16X4_F32 along K (exact f32).
// ---------------------------------------------------------------------------
__global__ void gemm_wmma_f32(const float* __restrict__ A, const float* __restrict__ B,
                              float* __restrict__ C, int M, int N, int K) {
    int wave = (int)((blockIdx.x * blockDim.x + threadIdx.x) >> 5);
    int lane = threadIdx.x & 31;
    int tilesN = N >> 4;
    int tiles = (M >> 4) * tilesN;
    if (wave >= tiles) return;                 // wave-uniform guard
    int tm   = wave / tilesN;
    int tn   = wave - tm * tilesN;
    int half = lane >> 4;                      // 0 or 1
    int l15  = lane & 15;
    int row  = tm * 16 + l15;                  // A row for this lane
    int col  = tn * 16 + l15;                  // B/C col for this lane
    int kh   = half * 2;                       // lane-half K offset
    const float* Ap = A + (size_t)row * K + kh;
    const float* Bp = B + (size_t)kh * N + col;
    v8f acc = {};
    for (int k = 0; k < K; k += 4) {
        v2f a, b;
        a.x = Ap[k];
        a.y = Ap[k + 1];
        b.x = Bp[(size_t)k * N];
        b.y = Bp[(size_t)(k + 1) * N];
        acc = __builtin_amdgcn_wmma_f32_16x16x4_f32(false, a, false, b,
                                                    (short)0, acc, false, false);
    }
    int rbase = tm * 16 + half * 8;
    float* Cp = C + (size_t)rbase * N + col;
#pragma unroll
    for (int i = 0; i < 8; i++) Cp[(size_t)i * N] = acc[i];
}

// ---------------------------------------------------------------------------
// alpha_s[n,h] = dot(h[n,h,:], a_src[h,:]); same for alpha_d.
// ---------------------------------------------------------------------------
__global__ void alpha_kernel(const float* __restrict__ h, const float* __restrict__ a_src,
                             const float* __restrict__ a_dst, float* __restrict__ as,
                             float* __restrict__ ad, int n, int H, int C) {
    int idx = blockIdx.x * blockDim.x + threadIdx.x;
    if (idx >= n * H) return;
    int node = idx / H, hd = idx - node * H;
    const float* hp = h + ((size_t)node * H + hd) * C;
    const float* sv = a_src + (size_t)hd * C;
    const float* dv = a_dst + (size_t)hd * C;
    float s = 0.f, d = 0.f;
    for (int c = 0; c < C; c++) { float v = hp[c]; s += v * sv[c]; d += v * dv[c]; }
    as[idx] = s; ad[idx] = d;
}

__device__ __forceinline__ void edge_sd(const int* __restrict__ ei, int e, int E0,
                                        int& s, int& d) {
    if (e < E0) { s = ei[e]; d = ei[E0 + e]; }
    else        { s = d = e - E0; }            // self-loop
}

__device__ __forceinline__ float leaky(float v) {
    return v > 0.f ? v : NEG_SLOPE * v;
}

// ---------------------------------------------------------------------------
// CSR-by-dst construction
// ---------------------------------------------------------------------------
__global__ void deg_kernel(const int* __restrict__ ei, unsigned int* __restrict__ deg,
                           int E0, int E) {
    int e = blockIdx.x * blockDim.x + threadIdx.x;
    if (e >= E) return;
    int s, d; edge_sd(ei, e, E0, s, d);
    atomicAdd(&deg[d], 1u);
}

// per-block exclusive scan (LDS Hillis-Steele), emits block totals
__global__ void scan_block(const unsigned int* __restrict__ in, int* __restrict__ outEx,
                           unsigned int* __restrict__ bsum, int n) {
    __shared__ unsigned int sh[256];
    int t = threadIdx.x;
    int i = blockIdx.x * 256 + t;
    unsigned int v = (i < n) ? in[i] : 0u;
    sh[t] = v;
    __syncthreads();
    for (int off = 1; off < 256; off <<= 1) {
        unsigned int x = (t >= off) ? sh[t - off] : 0u;
        __syncthreads();
        sh[t] += x;
        __syncthreads();
    }
    if (i < n) outEx[i] = (int)(sh[t] - v);
    if (t == 255) bsum[blockIdx.x] = sh[255];
}

// single-block exclusive scan over block totals (nb <= 256)
__global__ void scan_tops(const unsigned int* __restrict__ bsum,
                          unsigned int* __restrict__ bx, int nb) {
    __shared__ unsigned int sh[256];
    int t = threadIdx.x;
    unsigned int v = (t < nb) ? bsum[t] : 0u;
    sh[t] = v;
    __syncthreads();
    for (int off = 1; off < 256; off <<= 1) {
        unsigned int x = (t >= off) ? sh[t - off] : 0u;
        __syncthreads();
        sh[t] += x;
        __syncthreads();
    }
    if (t < nb) bx[t] = sh[t] - v;
}

__global__ void scan_add(int* __restrict__ rowptr, const unsigned int* __restrict__ bx,
                         int n, int E) {
    int i = blockIdx.x * blockDim.x + threadIdx.x;
    if (i == 0) rowptr[n] = E;
    if (i < n) rowptr[i] += (int)bx[i >> 8];
}

__global__ void fill_kernel(const int* __restrict__ ei, const int* __restrict__ rowptr,
                            unsigned int* __restrict__ cnt, int* __restrict__ csr_src,
                            int E0, int E) {
    int e = blockIdx.x * blockDim.x + threadIdx.x;
    if (e >= E) return;
    int s, d; edge_sd(ei, e, E0, s, d);
    unsigned int off = atomicAdd(&cnt[d], 1u);
    csr_src[rowptr[d] + (int)off] = s;
}

// ---------------------------------------------------------------------------
// Fused per-dst-node GAT softmax + gather-aggregation + bias(+ReLU).
// One wave per node; no float atomics anywhere.
//   phase1: segment max per head   (lane-strided + __shfl_xor wave32 reduce)
//   phase2: sum of exp per head    (same)
//   phase3: each lane owns CH=F/32 CONTIGUOUS channels -> unrolled adjacent
//           loads/stores merge into global_load/store_b128 (b64 for F=64)
// Compile-time H, C so the backend can merge and constant-fold head indices.
// ---------------------------------------------------------------------------
template <int H, int C>
__global__ void node_gat_kernel(const int* __restrict__ rowptr, const int* __restrict__ csr_src,
                                const float* __restrict__ as, const float* __restrict__ ad,
                                const float* __restrict__ hf, const float* __restrict__ bias,
                                float* __restrict__ out, int n, int do_relu) {
    constexpr int F  = H * C;
    constexpr int CH = F / 32;                 // contiguous channels per lane (8 or 2)
    int wave = (int)((blockIdx.x * blockDim.x + threadIdx.x) >> 5);
    int lane = threadIdx.x & 31;
    if (wave >= n) return;                     // wave-uniform guard
    int d = wave;
    int begin = rowptr[d], end = rowptr[d + 1];   // end > begin (self-loop)
    float adv[H];
#pragma unroll
    for (int h = 0; h < H; h++) adv[h] = ad[d * H + h];

    // phase 1: per-head segment max
    float mx[H];
#pragma unroll
    for (int h = 0; h < H; h++) mx[h] = -INFINITY;
    for (int j = begin + lane; j < end; j += 32) {
        int s = csr_src[j];
#pragma unroll
        for (int h = 0; h < H; h++)
            mx[h] = fmaxf(mx[h], leaky(as[s * H + h] + adv[h]));
    }
#pragma unroll
    for (int off = 16; off; off >>= 1)
#pragma unroll
        for (int h = 0; h < H; h++)
            mx[h] = fmaxf(mx[h], __shfl_xor(mx[h], off, 32));

    // phase 2: per-head sum of exp(e - max)
    float sm[H];
#pragma unroll
    for (int h = 0; h < H; h++) sm[h] = 0.f;
    for (int j = begin + lane; j < end; j += 32) {
        int s = csr_src[j];
#pragma unroll
        for (int h = 0; h < H; h++)
            sm[h] += expf(leaky(as[s * H + h] + adv[h]) - mx[h]);
    }
#pragma unroll
    for (int off = 16; off; off >>= 1)
#pragma unroll
        for (int h = 0; h < H; h++)
            sm[h] += __shfl_xor(sm[h], off, 32);
    float rden[H];
#pragma unroll
    for (int h = 0; h < H; h++) rden[h] = 1.f / (sm[h] + 1e-16f);

    // phase 3: gather-weighted aggregation; lane owns channels [lane*CH, lane*CH+CH)
    int c0 = lane * CH;
    float acc[CH];
#pragma unroll
    for (int i = 0; i < CH; i++) acc[i] = 0.f;
    for (int j = begin; j < end; j++) {
        int s = csr_src[j];                    // wave-uniform
        float al[H];
#pragma unroll
        for (int h = 0; h < H; h++)
            al[h] = expf(leaky(as[s * H + h] + adv[h]) - mx[h]) * rden[h];
        const float* hrow = hf + (size_t)s * F + c0;
#pragma unroll
        for (int i = 0; i < CH; i++)
            acc[i] += hrow[i] * al[(c0 + i) / C];    // adjacent loads merge to b128/b64
    }
    float* orow = out + (size_t)d * F + c0;
    const float* brow = bias + c0;
#pragma unroll
    for (int i = 0; i < CH; i++) {
        float v = acc[i] + brow[i];
        orow[i] = do_relu ? fmaxf(v, 0.f) : v;       // adjacent stores merge
    }
}

extern "C" void kernel_launch(void* const* d_in, const int* in_sizes, int n_in,
                              void* d_out, int out_size, void* d_ws, size_t ws_size,
                              hipStream_t stream) {
    (void)n_in; (void)ws_size; (void)out_size;
    const float* x      = (const float*)d_in[0];
    const int*   ei     = (const int*)d_in[1];     // edge_index [2,E0]
    const float* W1     = (const float*)d_in[2];
    const float* a_src1 = (const float*)d_in[3];
    const float* a_dst1 = (const float*)d_in[4];
    const float* b1     = (const float*)d_in[5];
    const float* W2     = (const float*)d_in[6];
    const float* a_src2 = (const float*)d_in[7];
    const float* a_dst2 = (const float*)d_in[8];
    const float* b2     = (const float*)d_in[9];

    const int Fin = 128, H1 = 4, C1 = 64, F1 = H1 * C1, C2 = 64;
    const int n  = in_sizes[0] / Fin;          // 50000
    const int E0 = in_sizes[1] / 2;            // 800000
    const int E  = E0 + n;                     // + self loops

    // ---- workspace carving ----
    float* ws = (float*)d_ws;
    float* h1  = ws;                  ws += (size_t)n * F1;   // layer1 linear out
    float* h2  = ws;                  ws += (size_t)n * F1;   // layer1 final (relu)
    float* as1 = ws;                  ws += (size_t)n * H1;
    float* ad1 = ws;                  ws += (size_t)n * H1;
    int* rowptr        = (int*)ws;          ws += (size_t)(n + 1);
    unsigned int* deg  = (unsigned int*)ws; ws += (size_t)n;
    unsigned int* cnt  = (unsigned int*)ws; ws += (size_t)n;
    int* csr_src       = (int*)ws;          ws += (size_t)E;
    unsigned int* bsum = (unsigned int*)ws; ws += 256;
    unsigned int* btop = (unsigned int*)ws; ws += 256;
    // layer-2 aliases (source regions dead by then)
    float* h2lin = h1;
    float* as2   = as1;
    float* ad2   = ad1;

    const int TB = 256;
    const int nb = (n + 255) / 256;            // scan blocks (<= 256)

    // ---- CSR-by-dst build (integer atomics only) ----
    hipMemsetAsync(deg, 0, (size_t)n * 4, stream);
    hipMemsetAsync(cnt, 0, (size_t)n * 4, stream);
    deg_kernel<<<(E + TB - 1) / TB, TB, 0, stream>>>(ei, deg, E0, E);
    scan_block<<<nb, 256, 0, stream>>>(deg, rowptr, bsum, n);
    scan_tops<<<1, 256, 0, stream>>>(bsum, btop, nb);
    scan_add<<<nb, 256, 0, stream>>>(rowptr, btop, n, E);
    fill_kernel<<<(E + TB - 1) / TB, TB, 0, stream>>>(ei, rowptr, cnt, csr_src, E0, E);

    // ---- layer 1 ----
    {
        int tiles = (n / 16) * (F1 / 16);
        gemm_wmma_f32<<<(tiles * 32 + TB - 1) / TB, TB, 0, stream>>>(x, W1, h1, n, F1, Fin);
    }
    alpha_kernel<<<(n * H1 + TB - 1) / TB, TB, 0, stream>>>(h1, a_src1, a_dst1, as1, ad1, n, H1, C1);
    node_gat_kernel<4, 64><<<(int)(((size_t)n * 32 + TB - 1) / TB), TB, 0, stream>>>(
        rowptr, csr_src, as1, ad1, h1, b1, h2, n, 1);

    // ---- layer 2 ----
    {
        int tiles = (n / 16) * (C2 / 16);
        gemm_wmma_f32<<<(tiles * 32 + TB - 1) / TB, TB, 0, stream>>>(h2, W2, h2lin, n, C2, F1);
    }
    alpha_kernel<<<(n + TB - 1) / TB, TB, 0, stream>>>(h2lin, a_src2, a_dst2, as2, ad2, n, 1, C2);
    node_gat_kernel<1, 64><<<(int)(((size_t)n * 32 + TB - 1) / TB), TB, 0, stream>>>(
        rowptr, csr_src, as2, ad2, h2lin, b2, (float*)d_out, n, 0);
}